// Model_25769804184
// MI455X (gfx1250) — compile-verified
//
#include <hip/hip_runtime.h>
#include <hip/hip_bf16.h>

typedef __attribute__((ext_vector_type(2))) float v2f;
typedef __attribute__((ext_vector_type(8))) float v8f;

#define TT 256
#define BB 256
#define HH 100
#define GG 400   // 4*H gates
#define EMBD 50
#define XP 52    // EMBD padded to multiple of 4
#define LBL 66
#define LBLP 80  // LBL padded to 5 col tiles
#define NBLK 100 // persistent grid: 100 blocks * 8 waves = 800 waves = tile jobs/step

__device__ __forceinline__ float sigmoidf_(float x) {
  return 1.0f / (1.0f + __expf(-x));
}

__device__ __forceinline__ v8f wmma4(v2f a, v2f b, v8f c) {
  return __builtin_amdgcn_wmma_f32_16x16x4_f32(false, a, false, b,
                                               (short)0, c, false, false);
}

// Device-wide split barrier over NBLK resident workgroups (persistent kernel).
__device__ __forceinline__ void grid_sync(volatile unsigned* counter,
                                          volatile unsigned* gen,
                                          unsigned nblocks) {
  __syncthreads();
  if (threadIdx.x == 0) {
    __threadfence();
    unsigned g = *gen;
    unsigned arrived = atomicAdd((unsigned*)counter, 1u);
    if (arrived == nblocks - 1u) {
      *counter = 0u;
      __threadfence();
      *gen = g + 1u;
    } else {
      while (*gen == g) { __builtin_amdgcn_s_sleep(1); }
      __threadfence();
    }
  }
  __syncthreads();
}

__global__ void init_barrier_kernel(unsigned* bs) {
  if (threadIdx.x < 2) bs[threadIdx.x] = 0u;
}

// x0[t][b][e] = emb_table[sentence[b][t]][e], K padded 50 -> 52 with zeros.
__global__ __launch_bounds__(256) void embed_kernel(const int* __restrict__ sent,
                                                    const float* __restrict__ table,
                                                    float* __restrict__ x0) {
  int i = blockIdx.x * 256 + threadIdx.x;
  if (i >= TT * BB * XP) return;
  int e = i % XP;
  int tb = i / XP;
  int b = tb % BB;
  int t = tb / BB;
  int idx = sent[b * TT + t];
  x0[i] = (e < EMBD) ? table[(size_t)idx * EMBD + e] : 0.0f;
}

// Pad w_ih_l0 (both dirs) from [400][50] to [2][400][52] with zeros.
__global__ __launch_bounds__(256) void pad_wih_kernel(const float* __restrict__ wf,
                                                      const float* __restrict__ wb,
                                                      float* __restrict__ wp) {
  int i = blockIdx.x * 256 + threadIdx.x;
  if (i >= 2 * GG * XP) return;
  int d = i / (GG * XP);
  int r = (i / XP) % GG;
  int k = i % XP;
  const float* w = d ? wb : wf;
  wp[i] = (k < EMBD) ? w[r * EMBD + k] : 0.0f;
}

// Pad w_out from [66][200] to [80][200] (zero rows) and b_out to [80].
__global__ __launch_bounds__(256) void pad_wout_kernel(const float* __restrict__ w,
                                                       const float* __restrict__ b,
                                                       float* __restrict__ wp,
                                                       float* __restrict__ bp) {
  int i = blockIdx.x * 256 + threadIdx.x;
  if (i < LBLP * 2 * HH) {
    int r = i / (2 * HH);
    int k = i % (2 * HH);
    wp[i] = (r < LBL) ? w[r * 2 * HH + k] : 0.0f;
  } else if (i < LBLP * 2 * HH + LBLP) {
    int r = i - LBLP * 2 * HH;
    bp[r] = (r < LBL) ? b[r] : 0.0f;
  }
}

// One persistent kernel runs the whole bidirectional scan of one layer.
// D is compile-time (52 or 200), all K loops fully unrolled and guard-free.
template <int D, bool PRELOAD_WIH>
__global__ __launch_bounds__(256) void lstm_layer_kernel(
    const float* __restrict__ x,
    const float* __restrict__ wih_f, const float* __restrict__ whh_f,
    const float* __restrict__ bias_f,
    const float* __restrict__ wih_b, const float* __restrict__ whh_b,
    const float* __restrict__ bias_b,
    float* __restrict__ hout,   // [T][B][2H]
    float* __restrict__ g,      // [2][B][GG] scratch
    float* __restrict__ c,      // [2][B][HH] scratch
    unsigned* __restrict__ bs)  // {counter, gen}
{
  constexpr int KI = D / 4;   // input-projection k-steps
  constexpr int KH = HH / 4;  // recurrent k-steps (25)
  const int lane = threadIdx.x & 31;
  const int wave = blockIdx.x * 8 + (threadIdx.x >> 5); // 0..799
  const int dir = wave / 400;
  const int tile = wave - dir * 400;
  const int mt = tile / 25;            // batch tile 0..15
  const int nt = tile - mt * 25;       // gate-col tile 0..24
  const int laneN = lane & 15;
  const int laneHalf = lane >> 4;      // 0: K={0,1}, 1: K={2,3}
  const int koff = 2 * laneHalf;
  const int bm = mt * 16 + laneN;      // batch row this lane loads for A
  const int nc = nt * 16 + laneN;      // gate col this lane loads for B
  const float* wihRow = (dir ? wih_b : wih_f) + (size_t)nc * D + koff;
  const float* whhRow = (dir ? whh_b : whh_f) + (size_t)nc * HH + koff;
  volatile unsigned* counter = bs;
  volatile unsigned* gen = bs + 1;
  const int tidg = blockIdx.x * 256 + threadIdx.x;

  // Hoist time-invariant B fragments into registers.
  v2f bi[PRELOAD_WIH ? KI : 1];
  if (PRELOAD_WIH) {
#pragma unroll
    for (int k = 0; k < KI; ++k) bi[k] = *(const v2f*)(wihRow + 4 * k);
  }
  v2f bh[KH];
#pragma unroll
  for (int k = 0; k < KH; ++k) bh[k] = *(const v2f*)(whhRow + 4 * k);

  for (int t = 0; t < TT; ++t) {
    const int tx = dir ? (TT - 1 - t) : t;  // time index in the input
    // ---------------- phase 1: tiled GEMM into g ----------------
    v8f acc0 = {}, acc1 = {};
    const float* xrow = x + ((size_t)tx * BB + bm) * D + koff;
    {
      int tn = dir ? (tx > 0 ? tx - 1 : 0) : (tx < TT - 1 ? tx + 1 : tx);
      __builtin_prefetch(x + ((size_t)tn * BB + bm) * D, 0, 1);
    }
#pragma unroll
    for (int k = 0; k < KI; ++k) {
      v2f a = *(const v2f*)(xrow + 4 * k);
      v2f b = PRELOAD_WIH ? bi[k] : *(const v2f*)(wihRow + 4 * k);
      if (k & 1) acc1 = wmma4(a, b, acc1);
      else       acc0 = wmma4(a, b, acc0);
    }
    if (t > 0) {  // h_0 == 0: skip recurrent segment at t==0
      const int tp = dir ? (tx + 1) : (t - 1);  // slot holding h_{t-1}
      const float* hrow = hout + ((size_t)tp * BB + bm) * (2 * HH) + dir * HH + koff;
#pragma unroll
      for (int k = 0; k < KH; ++k) {
        v2f a = *(const v2f*)(hrow + 4 * k);
        if (k & 1) acc1 = wmma4(a, bh[k], acc1);
        else       acc0 = wmma4(a, bh[k], acc0);
      }
    }
    v8f acc = acc0 + acc1;
#pragma unroll
    for (int j = 0; j < 8; ++j) {       // C layout: row = j + 8*laneHalf, col = laneN
      int row = mt * 16 + j + 8 * laneHalf;
      g[((size_t)dir * BB + row) * GG + nc] = acc[j];
    }
    grid_sync(counter, gen, NBLK);
    // ---------------- phase 2: gates + state update ----------------
#pragma unroll
    for (int it = tidg; it < 2 * BB * HH; it += NBLK * 256) {
      int d2 = it / (BB * HH);
      int rem = it - d2 * (BB * HH);
      int b2 = rem / HH;
      int j = rem - b2 * HH;
      const float* bv = d2 ? bias_b : bias_f;
      const float* gr = g + ((size_t)d2 * BB + b2) * GG;
      float gi = gr[j]          + bv[j];
      float gf = gr[HH + j]     + bv[HH + j];
      float gc = gr[2 * HH + j] + bv[2 * HH + j];
      float go = gr[3 * HH + j] + bv[3 * HH + j];
      size_t cidx = ((size_t)d2 * BB + b2) * HH + j;
      float cold = (t == 0) ? 0.0f : c[cidx];
      float cn = sigmoidf_(gf) * cold + sigmoidf_(gi) * tanhf(gc);
      float hn = sigmoidf_(go) * tanhf(cn);
      c[cidx] = cn;
      int tw = d2 ? (TT - 1 - t) : t;
      hout[((size_t)tw * BB + b2) * (2 * HH) + d2 * HH + j] = hn;
    }
    grid_sync(counter, gen, NBLK);
  }
}

// logits[b][t][l] = h1[t][b][:] . w_out[l][:] + b_out[l]; guard-free K loop
// (w_out padded to 80 rows), store guard only.
__global__ __launch_bounds__(256) void output_kernel(const float* __restrict__ h1,
                                                     const float* __restrict__ w_out,
                                                     const float* __restrict__ b_out,
                                                     float* __restrict__ out) {
  const int lane = threadIdx.x & 31;
  const int wave = blockIdx.x * 8 + (threadIdx.x >> 5); // 0..20479
  const int mtile = wave / 5;           // 4096 row tiles over r = b*T + t
  const int ntile = wave - mtile * 5;   // 5 col tiles (80 cols, 66 valid)
  const int laneN = lane & 15;
  const int laneHalf = lane >> 4;
  const int koff = 2 * laneHalf;
  const int r = mtile * 16 + laneN;     // output row this lane loads for A
  const int b = r >> 8;                 // T == 256
  const int t = r & 255;
  const float* arow = h1 + ((size_t)t * BB + b) * (2 * HH) + koff;
  const int nc = ntile * 16 + laneN;
  const float* brow = w_out + (size_t)nc * (2 * HH) + koff;
  v8f acc0 = {}, acc1 = {};
#pragma unroll
  for (int k = 0; k < 2 * HH / 4; ++k) {
    v2f a = *(const v2f*)(arow + 4 * k);
    v2f bb = *(const v2f*)(brow + 4 * k);
    if (k & 1) acc1 = wmma4(a, bb, acc1);
    else       acc0 = wmma4(a, bb, acc0);
  }
  v8f acc = acc0 + acc1;
  if (nc < LBL) {
    float bias = b_out[nc];
#pragma unroll
    for (int j = 0; j < 8; ++j) {
      int row = mtile * 16 + j + 8 * laneHalf;
      out[(size_t)row * LBL + nc] = acc[j] + bias;
    }
  }
}

extern "C" void kernel_launch(void* const* d_in, const int* in_sizes, int n_in,
                              void* d_out, int out_size, void* d_ws, size_t ws_size,
                              hipStream_t stream) {
  (void)in_sizes; (void)n_in; (void)out_size; (void)ws_size;
  const int*   sent     = (const int*)  d_in[0];
  const float* emb      = (const float*)d_in[1];
  const float* wih_l0_f = (const float*)d_in[2];
  const float* whh_l0_f = (const float*)d_in[3];
  const float* b_l0_f   = (const float*)d_in[4];
  const float* wih_l0_b = (const float*)d_in[5];
  const float* whh_l0_b = (const float*)d_in[6];
  const float* b_l0_b   = (const float*)d_in[7];
  const float* wih_l1_f = (const float*)d_in[8];
  const float* whh_l1_f = (const float*)d_in[9];
  const float* b_l1_f   = (const float*)d_in[10];
  const float* wih_l1_b = (const float*)d_in[11];
  const float* whh_l1_b = (const float*)d_in[12];
  const float* b_l1_b   = (const float*)d_in[13];
  const float* w_out    = (const float*)d_in[14];
  const float* b_out    = (const float*)d_in[15];
  float* out = (float*)d_out;

  char* ws = (char*)d_ws;
  unsigned* bsdev = (unsigned*)ws;                         // barrier: counter, gen
  float* x0    = (float*)(ws + 256);                       // [T][B][XP]
  float* h0    = x0    + (size_t)TT * BB * XP;             // [T][B][2H]
  float* h1    = h0    + (size_t)TT * BB * 2 * HH;         // [T][B][2H]
  float* gbuf  = h1    + (size_t)TT * BB * 2 * HH;         // [2][B][GG]
  float* cbuf  = gbuf  + (size_t)2 * BB * GG;              // [2][B][HH]
  float* wih0p = cbuf  + (size_t)2 * BB * HH;              // [2][GG][XP]
  float* woutp = wih0p + (size_t)2 * GG * XP;              // [LBLP][2H]
  float* boutp = woutp + (size_t)LBLP * 2 * HH;            // [LBLP]

  init_barrier_kernel<<<1, 32, 0, stream>>>(bsdev);

  embed_kernel<<<(TT * BB * XP + 255) / 256, 256, 0, stream>>>(sent, emb, x0);
  pad_wih_kernel<<<(2 * GG * XP + 255) / 256, 256, 0, stream>>>(wih_l0_f, wih_l0_b, wih0p);
  pad_wout_kernel<<<(LBLP * 2 * HH + LBLP + 255) / 256, 256, 0, stream>>>(w_out, b_out,
                                                                          woutp, boutp);

  lstm_layer_kernel<XP, true><<<NBLK, 256, 0, stream>>>(
      x0,
      wih0p, whh_l0_f, b_l0_f,
      wih0p + (size_t)GG * XP, whh_l0_b, b_l0_b,
      h0, gbuf, cbuf, bsdev);

  lstm_layer_kernel<2 * HH, false><<<NBLK, 256, 0, stream>>>(
      h0,
      wih_l1_f, whh_l1_f, b_l1_f,
      wih_l1_b, whh_l1_b, b_l1_b,
      h1, gbuf, cbuf, bsdev);

  output_kernel<<<(BB * TT / 16) * 5 / 8, 256, 0, stream>>>(h1, woutp, boutp, out);
}